// GraphEncoder_30451318128787
// MI455X (gfx1250) — compile-verified
//
#include <hip/hip_runtime.h>

#define IN_F   128
#define HID_F  128
#define OUT_F  64
#define N_GR   32

typedef float v2f __attribute__((ext_vector_type(2)));
typedef float v8f __attribute__((ext_vector_type(8)));

// ---------------------------------------------------------------------------
// Init: deg = 1.0 (self-loop pre-counted), agg/psum/pcnt = 0
// ---------------------------------------------------------------------------
__global__ void init_kernel(float* __restrict__ deg, float* __restrict__ agg,
                            float* __restrict__ psum, float* __restrict__ pcnt,
                            int N) {
    long long i = (long long)blockIdx.x * blockDim.x + threadIdx.x;
    long long tot = (long long)N * HID_F;
    if (i < tot) agg[i] = 0.0f;
    if (i < N) deg[i] = 1.0f;
    if (i < N_GR * HID_F) psum[i] = 0.0f;
    if (i < N_GR) pcnt[i] = 0.0f;
}

// ---------------------------------------------------------------------------
// Degree: one atomic per edge onto dst
// ---------------------------------------------------------------------------
__global__ void degree_kernel(const long long* __restrict__ ei,
                              float* __restrict__ deg, int E) {
    int e = blockIdx.x * blockDim.x + threadIdx.x;
    if (e < E) atomicAdd(&deg[(int)ei[(size_t)E + e]], 1.0f);
}

// dinv = rsqrt(deg) in place (deg >= 1 always due to self-loops)
__global__ void dinv_kernel(float* __restrict__ deg, int N) {
    int i = blockIdx.x * blockDim.x + threadIdx.x;
    if (i < N) deg[i] = rsqrtf(deg[i]);
}

// ---------------------------------------------------------------------------
// GEMM  h[N,128] = x[N,128] @ W[128,128]  via V_WMMA_F32_16X16X4_F32
// One wave -> 16 rows x 128 cols strip; 8 v8f accumulators.
// A layout (32-bit 16x4, ISA 7.12.2): lanes 0-15 K={k,k+1}, lanes 16-31 K={k+2,k+3}
// B layout (4x16 rows striped across lanes): VGPR0 = row k (+2 for hi half),
//                                            VGPR1 = row k+1 (+2 for hi half)
// ---------------------------------------------------------------------------
__global__ __launch_bounds__(256) void gemm_kernel(const float* __restrict__ x,
                                                   const float* __restrict__ W,
                                                   float* __restrict__ h, int N) {
    const int wave = threadIdx.x >> 5;               // 0..7
    const int lane = threadIdx.x & 31;
    const int m0   = (blockIdx.x * 8 + wave) * 16;   // wave-uniform row tile
    if (m0 >= N) return;                             // uniform exit: EXEC stays all-1s

    const int half = lane >> 4;                      // 0 = lanes 0-15, 1 = lanes 16-31
    const int l15  = lane & 15;

    int row = m0 + l15;
    if (row >= N) row = N - 1;                       // clamp; store is masked below
    const float* xrow = x + (size_t)row * IN_F;

    v8f acc[8];
    v8f zero = {};
    #pragma unroll
    for (int n = 0; n < 8; ++n) acc[n] = zero;

    for (int k = 0; k < IN_F; k += 4) {
        // A fragment: contiguous float2 per lane
        const v2f a = *(const v2f*)(xrow + k + 2 * half);
        const float* wp = W + (size_t)(k + 2 * half) * HID_F + l15;
        #pragma unroll
        for (int n = 0; n < 8; ++n) {
            v2f b;
            b.x = wp[n * 16];               // row k+2*half
            b.y = wp[n * 16 + HID_F];       // row k+2*half+1
            acc[n] = __builtin_amdgcn_wmma_f32_16x16x4_f32(
                false, a, false, b, (short)0, acc[n], false, false);
        }
    }

    // C/D layout: VGPR v -> (M = m0 + v + 8*half, N = n*16 + l15)
    #pragma unroll
    for (int n = 0; n < 8; ++n) {
        #pragma unroll
        for (int v = 0; v < 8; ++v) {
            int r = m0 + v + 8 * half;
            if (r < N) h[(size_t)r * HID_F + n * 16 + l15] = acc[n][v];
        }
    }
}

// ---------------------------------------------------------------------------
// Edge scatter: one wave per edge, float4 of channels per lane,
// agg[dst] += h[src] * dinv[src]*dinv[dst]
// ---------------------------------------------------------------------------
__global__ __launch_bounds__(256) void edge_kernel(const long long* __restrict__ ei,
                                                   const float* __restrict__ h,
                                                   const float* __restrict__ dinv,
                                                   float* __restrict__ agg, int E) {
    const int lane = threadIdx.x & 31;
    const long long e = (long long)blockIdx.x * 8 + (threadIdx.x >> 5);
    if (e >= E) return;
    const int s = (int)ei[e];
    const int d = (int)ei[(size_t)E + e];
    const float w = dinv[s] * dinv[d];
    const float4 v = *((const float4*)(h + (size_t)s * HID_F) + lane);
    float* ap = agg + (size_t)d * HID_F + lane * 4;
    atomicAdd(ap + 0, v.x * w);
    atomicAdd(ap + 1, v.y * w);
    atomicAdd(ap + 2, v.z * w);
    atomicAdd(ap + 3, v.w * w);
}

// ---------------------------------------------------------------------------
// Fused self-loop + bias + ReLU + segmented mean-pool accumulation.
// batch is sorted, so a block of 32 nodes spans ~1-2 graphs: accumulate in a
// register, flush with one atomic per (graph-run, channel).
// ---------------------------------------------------------------------------
#define NODES_PER_BLK 32
__global__ __launch_bounds__(128) void pool_kernel(const float* __restrict__ agg,
                                                   const float* __restrict__ h,
                                                   const float* __restrict__ dinv,
                                                   const float* __restrict__ b_gcn,
                                                   const long long* __restrict__ batch,
                                                   float* __restrict__ psum,
                                                   float* __restrict__ pcnt, int N) {
    const int c = threadIdx.x;                     // channel 0..127
    const int base = blockIdx.x * NODES_PER_BLK;
    const float bias = b_gcn[c];

    float accv = 0.0f, cntv = 0.0f;
    int cur_g = -1;
    for (int i = 0; i < NODES_PER_BLK; ++i) {
        const int node = base + i;
        if (node >= N) break;
        const int g = (int)batch[node];
        if (g != cur_g) {
            if (cur_g >= 0) {
                atomicAdd(&psum[cur_g * HID_F + c], accv);
                if (c == 0) atomicAdd(&pcnt[cur_g], cntv);
            }
            cur_g = g; accv = 0.0f; cntv = 0.0f;
        }
        const float di = dinv[node];
        const float val = agg[(size_t)node * HID_F + c]
                        + h[(size_t)node * HID_F + c] * di * di   // self-loop msg
                        + bias;
        accv += fmaxf(val, 0.0f);                  // ReLU
        cntv += 1.0f;
    }
    if (cur_g >= 0) {
        atomicAdd(&psum[cur_g * HID_F + c], accv);
        if (c == 0) atomicAdd(&pcnt[cur_g], cntv);
    }
}

// ---------------------------------------------------------------------------
// out[32,64] = tanh((psum/cnt) @ W_lin + b_lin)  (tiny; scalar fp32)
// ---------------------------------------------------------------------------
__global__ void final_kernel(const float* __restrict__ psum,
                             const float* __restrict__ pcnt,
                             const float* __restrict__ W_lin,
                             const float* __restrict__ b_lin,
                             float* __restrict__ out) {
    const int t = blockIdx.x * blockDim.x + threadIdx.x;
    if (t >= N_GR * OUT_F) return;
    const int g = t >> 6, o = t & (OUT_F - 1);
    const float inv = 1.0f / fmaxf(pcnt[g], 1.0f);
    float s = b_lin[o];
    for (int k = 0; k < HID_F; ++k)
        s += (psum[g * HID_F + k] * inv) * W_lin[k * OUT_F + o];
    out[t] = tanhf(s);
}

// ---------------------------------------------------------------------------
extern "C" void kernel_launch(void* const* d_in, const int* in_sizes, int n_in,
                              void* d_out, int out_size, void* d_ws, size_t ws_size,
                              hipStream_t stream) {
    const float*     x      = (const float*)d_in[0];
    const long long* ei     = (const long long*)d_in[1];
    const long long* batch  = (const long long*)d_in[2];
    const float*     W_gcn  = (const float*)d_in[3];
    const float*     b_gcn  = (const float*)d_in[4];
    const float*     W_lin  = (const float*)d_in[5];
    const float*     b_lin  = (const float*)d_in[6];
    float*           out    = (float*)d_out;

    const int N = in_sizes[0] / IN_F;
    const int E = in_sizes[1] / 2;

    char* ws = (char*)d_ws;
    size_t off = 0;
    auto alloc = [&](size_t nbytes) -> void* {
        void* p = (void*)(ws + off);
        off += (nbytes + 255) & ~(size_t)255;
        return p;
    };
    float* deg  = (float*)alloc((size_t)N * sizeof(float));          // becomes dinv
    float* h    = (float*)alloc((size_t)N * HID_F * sizeof(float));
    float* agg  = (float*)alloc((size_t)N * HID_F * sizeof(float));
    float* psum = (float*)alloc((size_t)N_GR * HID_F * sizeof(float));
    float* pcnt = (float*)alloc((size_t)N_GR * sizeof(float));

    const long long tot = (long long)N * HID_F;
    init_kernel  <<<(int)((tot + 255) / 256), 256, 0, stream>>>(deg, agg, psum, pcnt, N);
    degree_kernel<<<(E + 255) / 256,          256, 0, stream>>>(ei, deg, E);
    dinv_kernel  <<<(N + 255) / 256,          256, 0, stream>>>(deg, N);
    gemm_kernel  <<<(N + 127) / 128,          256, 0, stream>>>(x, W_gcn, h, N);
    edge_kernel  <<<(E + 7) / 8,              256, 0, stream>>>(ei, h, deg, agg, E);
    pool_kernel  <<<(N + NODES_PER_BLK - 1) / NODES_PER_BLK, 128, 0, stream>>>(
                     agg, h, deg, b_gcn, batch, psum, pcnt, N);
    final_kernel <<<(N_GR * OUT_F + 255) / 256, 256, 0, stream>>>(psum, pcnt, W_lin, b_lin, out);
}